// GaussianKDE_82635170775443
// MI455X (gfx1250) — compile-verified
//
#include <hip/hip_runtime.h>
#include <cmath>

typedef __attribute__((ext_vector_type(2))) float v2f;
typedef __attribute__((ext_vector_type(8))) float v8f;

#define KD 32            // feature dimension
#define WAVES 8          // waves per block (256 threads, wave32)
#define XT 4             // X tiles (of 16 rows) per workgroup: 4x B-operand reuse

// ---- pre-kernel: ev[i] = exp(-0.5 * ||P_i||^2) -----------------------------
__global__ __launch_bounds__(256)
void row_norm_exp_kernel(const float* __restrict__ P, float* __restrict__ ev,
                         int rows) {
    int i = blockIdx.x * 256 + threadIdx.x;
    if (i >= rows) return;
    const float4* p = (const float4*)(P + (size_t)i * KD);
    float s = 0.0f;
#pragma unroll
    for (int k = 0; k < KD / 4; ++k) {
        float4 v = p[k];
        s = fmaf(v.x, v.x, s);
        s = fmaf(v.y, v.y, s);
        s = fmaf(v.z, v.z, s);
        s = fmaf(v.w, v.w, s);
    }
    ev[i] = __expf(-0.5f * s);
}

// ---- main: out[m] = scale * ex[m] * sum_j exp(x_m . y_j) * ey[j] -----------
__global__ __launch_bounds__(256)
void kde_wmma_kernel(const float* __restrict__ X, const float* __restrict__ Y,
                     const float* __restrict__ EX, const float* __restrict__ EY,
                     float* __restrict__ out, int Mrows, float scale) {
    __shared__ float partial[WAVES][XT][16];

    const int lane  = threadIdx.x & 31;
    const int wave  = threadIdx.x >> 5;
    const int xg    = blockIdx.x;         // group of XT 16-row X tiles
    const int row   = lane & 15;          // A-frag row m / B-frag col n for this lane
    const int khalf = lane >> 4;          // which K pair within each group of 4

    // A fragments for XT X tiles, loaded once:
    // lane holds X[tile_t, row][4k + 2*khalf + {0,1}]
    v2f a[XT][8];
#pragma unroll
    for (int t = 0; t < XT; ++t) {
        const float* xp =
            X + (size_t)((xg * XT + t) * 16 + row) * KD + 2 * khalf;
#pragma unroll
        for (int k = 0; k < 8; ++k)
            a[t][k] = *(const v2f*)(xp + 4 * k);
    }

    v8f acc[XT];
#pragma unroll
    for (int t = 0; t < XT; ++t) acc[t] = (v8f){};

    const int ytiles = Mrows / 16;
    for (int jt = wave; jt < ytiles; jt += WAVES) {
        // B fragments (B = Y^T): identical element indexing into Y
        const float* yp = Y + (size_t)(jt * 16 + row) * KD + 2 * khalf;
        v2f b[8];
#pragma unroll
        for (int k = 0; k < 8; ++k)
            b[k] = *(const v2f*)(yp + 4 * k);
        // exp(-0.5*||y_n||^2) for this lane's C column n = row
        const float ey = EY[jt * 16 + row];

        // XT independent WMMA chains, interleaved k-major so 4 independent
        // WMMAs separate each dependent D->C accumulation
        v8f c[XT];
#pragma unroll
        for (int t = 0; t < XT; ++t) c[t] = (v8f){};
#pragma unroll
        for (int k = 0; k < 8; ++k) {
#pragma unroll
            for (int t = 0; t < XT; ++t) {
                c[t] = __builtin_amdgcn_wmma_f32_16x16x4_f32(
                        /*neg_a=*/false, a[t][k], /*neg_b=*/false, b[k],
                        /*c_mod=*/(short)0, c[t],
                        /*reuse_a=*/false, /*reuse_b=*/false);
            }
        }

        // fused epilogue: acc += exp(dot) * ey
#pragma unroll
        for (int t = 0; t < XT; ++t)
#pragma unroll
            for (int v = 0; v < 8; ++v)
                acc[t][v] = fmaf(__expf(c[t][v]), ey, acc[t][v]);
    }

    // reduce over the 16 N-columns (each half-wave shares its m-group)
#pragma unroll
    for (int off = 1; off < 16; off <<= 1) {
#pragma unroll
        for (int t = 0; t < XT; ++t)
#pragma unroll
            for (int v = 0; v < 8; ++v)
                acc[t][v] += __shfl_xor(acc[t][v], off, 32);
    }

    // lane 0 holds row sums for m=0..7, lane 16 for m=8..15 (per tile)
    if (row == 0) {
#pragma unroll
        for (int t = 0; t < XT; ++t)
#pragma unroll
            for (int v = 0; v < 8; ++v)
                partial[wave][t][8 * khalf + v] = acc[t][v];
    }
    __syncthreads();

    // deterministic cross-wave combine + final scaling by exp(-0.5||x||^2)
    if (threadIdx.x < 16 * XT) {
        const int t = threadIdx.x >> 4;
        const int r = threadIdx.x & 15;
        float s = 0.0f;
#pragma unroll
        for (int w = 0; w < WAVES; ++w)
            s += partial[w][t][r];
        int m = (xg * XT + t) * 16 + r;
        out[m] = s * scale * EX[m];
    }
}

extern "C" void kernel_launch(void* const* d_in, const int* in_sizes, int n_in,
                              void* d_out, int out_size, void* d_ws, size_t ws_size,
                              hipStream_t stream) {
    const float* X = (const float*)d_in[0];
    const float* Y = (const float*)d_in[1];
    float* out = (float*)d_out;

    const int N = in_sizes[0] / KD;   // 16384
    const int M = in_sizes[1] / KD;   // 16384

    // scratch layout: ey[M] then ex[N]
    float* EY = (float*)d_ws;
    float* EX = EY + M;

    // exp(logp) = exp(-0.5*sq) * (2*pi)^(-d/2); fold constant and 1/N into scale
    const float scale =
        (float)(std::exp(-0.5 * (double)KD * std::log(2.0 * M_PI)) / (double)N);

    hipLaunchKernelGGL(row_norm_exp_kernel, dim3((M + 255) / 256), dim3(256), 0,
                       stream, Y, EY, M);
    hipLaunchKernelGGL(row_norm_exp_kernel, dim3((N + 255) / 256), dim3(256), 0,
                       stream, X, EX, N);
    hipLaunchKernelGGL(kde_wmma_kernel, dim3(N / (16 * XT)), dim3(256), 0,
                       stream, X, Y, EX, EY, out, M, scale);
}